// GraphEncoder_84559316124099
// MI455X (gfx1250) — compile-verified
//
#include <hip/hip_runtime.h>

#define N_NODES 20000
#define N_EDGES 320000
#define IN_DIM  128
#define HID_DIM 256
#define OUT_DIM 256   // HID_DIM == OUT_DIM == 256: helpers hardcode 256 cols

typedef float v2f __attribute__((ext_vector_type(2)));
typedef float v8f __attribute__((ext_vector_type(8)));

// ---------------- degree / norm ----------------

__global__ void k_init_deg(float* __restrict__ deg) {
    int i = blockIdx.x * blockDim.x + threadIdx.x;
    if (i < N_NODES) deg[i] = 1.0f;   // self-loop
}

__global__ void k_deg_accum(const long long* __restrict__ dst, float* __restrict__ deg) {
    int e = blockIdx.x * blockDim.x + threadIdx.x;
    if (e < N_EDGES) unsafeAtomicAdd(&deg[(int)dst[e]], 1.0f);
}

__global__ void k_deg_to_dinv(float* __restrict__ deg) {
    int i = blockIdx.x * blockDim.x + threadIdx.x;
    if (i < N_NODES) {
        float d = deg[i];
        deg[i] = (d > 0.0f) ? rsqrtf(d) : 0.0f;
    }
}

// ---------------- fp32 WMMA GEMM: C[M,N] = A[M,K] @ B[K,N] ----------------
// One wave per 16x16 output tile; V_WMMA_F32_16X16X4_F32, K stepped by 4.
// A-frag (16x4): lane l holds A[l&15][2*(l>>4)+r] in vgpr r.
// B-frag (4x16): lane l holds B[2*(l>>4)+r][l&15] in vgpr r.
// C/D (16x16):   vgpr r holds row r + 8*(l>>4), col l&15.

__global__ void k_gemm_wmma_f32(const float* __restrict__ A, const float* __restrict__ B,
                                float* __restrict__ C, int M, int K, int N) {
    const int lane  = threadIdx.x & 31;
    const int wave  = blockIdx.x * (blockDim.x >> 5) + (threadIdx.x >> 5);
    const int ntiles = N >> 4;
    const int mt = wave / ntiles;
    const int nt = wave - mt * ntiles;
    if (mt * 16 >= M) return;                 // wave-uniform: EXEC stays all-ones

    const int mn    = lane & 15;
    const int khalf = (lane >> 4) << 1;       // 0 or 2

    const float* arow = A + (size_t)(mt * 16 + mn) * K;
    const float* bcol = B + (size_t)(nt * 16 + mn);

    v8f acc = {};
#pragma unroll 4
    for (int k = 0; k < K; k += 4) {
        const float2 av = *(const float2*)(arow + k + khalf);   // 8B-aligned
        v2f a; a.x = av.x; a.y = av.y;
        v2f b;
        b.x = bcol[(size_t)(k + khalf)     * N];
        b.y = bcol[(size_t)(k + khalf + 1) * N];
        acc = __builtin_amdgcn_wmma_f32_16x16x4_f32(
            /*neg_a=*/false, a, /*neg_b=*/false, b,
            /*c_mod=*/(short)0, acc, /*reuse_a=*/false, /*reuse_b=*/false);
    }

    const int row0 = mt * 16 + ((lane >> 4) << 3);
    float* crow = C + (size_t)row0 * N + (size_t)(nt * 16 + mn);
#pragma unroll
    for (int r = 0; r < 8; ++r) crow[(size_t)r * N] = acc[r];
}

// ---------------- self-loop init (no atomics): out = h*dinv^2 (+ bias) ----------------

__global__ void k_self_init(const float* __restrict__ h, const float* __restrict__ dinv,
                            const float* __restrict__ bias, int has_bias,
                            float* __restrict__ out) {
    int idx = blockIdx.x * blockDim.x + threadIdx.x;    // N_NODES * 64 float4 slots
    int node = idx >> 6;
    if (node >= N_NODES) return;
    int c4 = (idx & 63) << 2;
    float di = dinv[node];
    float s  = di * di;
    float4 hv = *(const float4*)(h + (size_t)node * 256 + c4);
    float4 o;
    o.x = hv.x * s; o.y = hv.y * s; o.z = hv.z * s; o.w = hv.w * s;
    if (has_bias) {
        float4 b = *(const float4*)(bias + c4);
        o.x += b.x; o.y += b.y; o.z += b.z; o.w += b.w;
    }
    *(float4*)(out + (size_t)node * 256 + c4) = o;
}

// ---------------- edge scatter: out[dst] += h[src] * dinv[src]*dinv[dst] ----------------
// 64 threads per edge, float4 gather + 4 native global_atomic_add_f32 each.

__global__ void k_edge_scatter(const long long* __restrict__ src, const long long* __restrict__ dst,
                               const float* __restrict__ dinv, const float* __restrict__ h,
                               float* __restrict__ out) {
    int idx = blockIdx.x * blockDim.x + threadIdx.x;    // N_EDGES * 64
    int e = idx >> 6;
    if (e >= N_EDGES) return;
    int s = (int)src[e];
    int d = (int)dst[e];
    float norm = dinv[s] * dinv[d];
    int c4 = (idx & 63) << 2;
    float4 hv = *(const float4*)(h + (size_t)s * 256 + c4);
    float* o = out + (size_t)d * 256 + c4;
    unsafeAtomicAdd(o + 0, hv.x * norm);
    unsafeAtomicAdd(o + 1, hv.y * norm);
    unsafeAtomicAdd(o + 2, hv.z * norm);
    unsafeAtomicAdd(o + 3, hv.w * norm);
}

// ---------------- bias + ReLU (in place) ----------------

__global__ void k_bias_relu(float* __restrict__ buf, const float* __restrict__ bias) {
    int idx = blockIdx.x * blockDim.x + threadIdx.x;    // N_NODES * 64
    int node = idx >> 6;
    if (node >= N_NODES) return;
    int c4 = (idx & 63) << 2;
    float4 v = *(float4*)(buf + (size_t)node * 256 + c4);
    float4 b = *(const float4*)(bias + c4);
    v.x = fmaxf(v.x + b.x, 0.0f);
    v.y = fmaxf(v.y + b.y, 0.0f);
    v.z = fmaxf(v.z + b.z, 0.0f);
    v.w = fmaxf(v.w + b.w, 0.0f);
    *(float4*)(buf + (size_t)node * 256 + c4) = v;
}

// ---------------- launcher ----------------

extern "C" void kernel_launch(void* const* d_in, const int* in_sizes, int n_in,
                              void* d_out, int out_size, void* d_ws, size_t ws_size,
                              hipStream_t stream) {
    (void)in_sizes; (void)n_in; (void)out_size; (void)ws_size;

    const float*     x   = (const float*)d_in[0];
    const long long* ei  = (const long long*)d_in[1];   // int64 [2, N_EDGES]
    const float*     W1  = (const float*)d_in[2];
    const float*     b1  = (const float*)d_in[3];
    const float*     W2  = (const float*)d_in[4];
    const float*     b2  = (const float*)d_in[5];
    float*           out = (float*)d_out;

    const long long* src = ei;             // edge_index[0]
    const long long* dst = ei + N_EDGES;   // edge_index[1]

    // workspace: dinv (20000 f32) | buf0 (20000x256 f32) | buf1 (20000x256 f32)
    char*  ws   = (char*)d_ws;
    float* dinv = (float*)ws;
    size_t off  = ((size_t)N_NODES * 4 + 255) & ~(size_t)255;
    float* buf0 = (float*)(ws + off);
    float* buf1 = buf0 + (size_t)N_NODES * 256;

    const int TB = 256;
    const int gNode   = (N_NODES + TB - 1) / TB;
    const int gEdge   = (N_EDGES + TB - 1) / TB;
    const int gNode64 = (N_NODES * 64 + TB - 1) / TB;   // 5000
    const int gEdge64 = (N_EDGES * 64 + TB - 1) / TB;   // 80000
    const int gGemm   = ((N_NODES / 16) * (256 / 16) * 32) / TB;  // 2500 blocks, 1 wave/tile

    // degree -> dinv (in place)
    k_init_deg   <<<gNode, TB, 0, stream>>>(dinv);
    k_deg_accum  <<<gEdge, TB, 0, stream>>>(dst, dinv);
    k_deg_to_dinv<<<gNode, TB, 0, stream>>>(dinv);

    // layer 1: h0 = x @ W1 ; agg = scatter(norm * h0) ; h = relu(agg + b1)
    k_gemm_wmma_f32<<<gGemm, TB, 0, stream>>>(x, W1, buf0, N_NODES, IN_DIM, HID_DIM);
    k_self_init    <<<gNode64, TB, 0, stream>>>(buf0, dinv, nullptr, 0, buf1);
    k_edge_scatter <<<gEdge64, TB, 0, stream>>>(src, dst, dinv, buf0, buf1);
    k_bias_relu    <<<gNode64, TB, 0, stream>>>(buf1, b1);

    // layer 2: h2 = h @ W2 ; out = scatter(norm * h2) + b2
    k_gemm_wmma_f32<<<gGemm, TB, 0, stream>>>(buf1, W2, buf0, N_NODES, HID_DIM, OUT_DIM);
    k_self_init    <<<gNode64, TB, 0, stream>>>(buf0, dinv, b2, 1, out);
    k_edge_scatter <<<gEdge64, TB, 0, stream>>>(src, dst, dinv, buf0, out);
}